// NaiveGCNClassifier_68204080660733
// MI455X (gfx1250) — compile-verified
//
#include <hip/hip_runtime.h>
#include <hip/hip_bf16.h>

typedef __attribute__((ext_vector_type(2))) float v2f;
typedef __attribute__((ext_vector_type(8))) float v8f;

#define D 64  // D_IN == D_HID == 64

// ---------------------------------------------------------------------------
// Degree accumulation: deg_out[src[e]] += 1, deg_in[dst[e]] += 1
// ---------------------------------------------------------------------------
__global__ void gcn_degrees(const int* __restrict__ src, const int* __restrict__ dst,
                            float* __restrict__ degout, float* __restrict__ degin, int nE) {
    int e = blockIdx.x * blockDim.x + threadIdx.x;
    if (e < nE) {
        atomicAdd(&degout[src[e]], 1.0f);
        atomicAdd(&degin[dst[e]], 1.0f);
    }
}

// In-place: deg -> clamp(deg,1)^-0.5 for both arrays
__global__ void gcn_norms(float* __restrict__ a, float* __restrict__ b, int n) {
    int i = blockIdx.x * blockDim.x + threadIdx.x;
    if (i < n) {
        a[i] = rsqrtf(fmaxf(a[i], 1.0f));
        b[i] = rsqrtf(fmaxf(b[i], 1.0f));
    }
}

// y[i,:] = x[i,:] * s[i]   (float4 per thread, 16 threads per row)
__global__ void gcn_rowscale(const float* __restrict__ x, const float* __restrict__ s,
                             float* __restrict__ y, int n) {
    long long g = (long long)blockIdx.x * blockDim.x + threadIdx.x;
    if (g >= (long long)n * 16) return;
    int i  = (int)(g >> 4);
    int f4 = (int)(g & 15) * 4;
    float sc = s[i];
    float4 v = *(const float4*)(x + (size_t)i * D + f4);
    v.x *= sc; v.y *= sc; v.z *= sc; v.w *= sc;
    *(float4*)(y + (size_t)i * D + f4) = v;
}

// agg[dst[e],:] += xs[src[e],:]   (float4 gather + 4 f32 atomics per thread)
__global__ void gcn_scatter(const float* __restrict__ xs, const int* __restrict__ src,
                            const int* __restrict__ dst, float* __restrict__ agg, int nE) {
    long long g = (long long)blockIdx.x * blockDim.x + threadIdx.x;
    if (g >= (long long)nE * 16) return;
    int e  = (int)(g >> 4);
    int f4 = (int)(g & 15) * 4;
    int s = src[e];
    int d = dst[e];
    float4 v = *(const float4*)(xs + (size_t)s * D + f4);
    float* o = agg + (size_t)d * D + f4;
    atomicAdd(o + 0, v.x);
    atomicAdd(o + 1, v.y);
    atomicAdd(o + 2, v.z);
    atomicAdd(o + 3, v.w);
}

// ---------------------------------------------------------------------------
// Fused dense layer: Y = relu( diag(prescale) * X @ W + bias ) * diag(postscale)
// X: [N,64] f32, W: [64,64] f32 (row-major), Y: [N,64].
// 128 threads = 4 waves; wave w handles rows [ (blk*4+w)*16 , +16 ).
// Uses V_WMMA_F32_16X16X4_F32, K tiled in 16 steps of 4.
// ISA f32 layouts (CDNA5 ISA §7.12.2):
//   A 16x4 : lanes 0-15 -> M=lane, v0=K0 v1=K1 ; lanes 16-31 -> M=lane-16, v0=K2 v1=K3
//   B 4x16 : lanes 0-15 -> N=lane, v0=K0 v1=K1 ; lanes 16-31 -> N=lane-16, v0=K2 v1=K3
//   C/D    : vgpr r -> M=r (lanes 0-15) / M=r+8 (lanes 16-31), N = lane&15
// ---------------------------------------------------------------------------
__global__ __launch_bounds__(128) void gcn_gemm64_wmma(
    const float* __restrict__ X, const float* __restrict__ W,
    const float* __restrict__ bias,
    const float* __restrict__ prescale,   // norm_dst, length N
    const float* __restrict__ postscale,  // norm_src or nullptr
    float* __restrict__ Y, int N) {
    __shared__ float Wlds[D * D];  // 16 KB

    int tid = threadIdx.x;
    // cooperative stage of W into LDS (4096 floats / 128 threads = 8 float4 each)
    for (int i = tid; i < D * D / 4; i += 128) {
        ((float4*)Wlds)[i] = ((const float4*)W)[i];
    }
    __syncthreads();

    int wave    = tid >> 5;
    int lane    = tid & 31;
    int half    = lane >> 4;   // 0: K0/K1 side, 1: K2/K3 side
    int l15     = lane & 15;
    int m0      = (blockIdx.x * 4 + wave) * 16;
    int row     = m0 + l15;
    int rowc    = row < N ? row : 0;              // clamped (loads stay in range)
    float rs    = (row < N) ? prescale[row] : 0.0f;  // OOB rows contribute zeros

    const float* xrow = X + (size_t)rowc * D;

    v8f acc[4] = {v8f{0.f,0.f,0.f,0.f,0.f,0.f,0.f,0.f},
                  v8f{0.f,0.f,0.f,0.f,0.f,0.f,0.f,0.f},
                  v8f{0.f,0.f,0.f,0.f,0.f,0.f,0.f,0.f},
                  v8f{0.f,0.f,0.f,0.f,0.f,0.f,0.f,0.f}};

    #pragma unroll
    for (int kk = 0; kk < 16; ++kk) {
        int kbase = kk * 4 + half * 2;
        v2f a;
        a.x = xrow[kbase + 0] * rs;
        a.y = xrow[kbase + 1] * rs;
        #pragma unroll
        for (int j = 0; j < 4; ++j) {
            v2f b;
            b.x = Wlds[(kbase + 0) * D + j * 16 + l15];
            b.y = Wlds[(kbase + 1) * D + j * 16 + l15];
            acc[j] = __builtin_amdgcn_wmma_f32_16x16x4_f32(
                false, a, false, b, (short)0, acc[j], false, false);
        }
    }

    // epilogue: bias + relu + optional postscale
    #pragma unroll
    for (int j = 0; j < 4; ++j) {
        int ocol = j * 16 + l15;
        float bv = bias[ocol];
        #pragma unroll
        for (int r = 0; r < 8; ++r) {
            int orow = m0 + r + half * 8;
            if (orow < N) {
                float v = fmaxf(acc[j][r] + bv, 0.0f);
                if (postscale) v *= postscale[orow];
                Y[(size_t)orow * D + ocol] = v;
            }
        }
    }
}

// pooled[gid[i],:] += h2[i,:] ; counts[gid[i]] += 1
__global__ void gcn_pool(const float* __restrict__ h2, const int* __restrict__ gid,
                         float* __restrict__ pooled, float* __restrict__ counts, int n) {
    long long g = (long long)blockIdx.x * blockDim.x + threadIdx.x;
    if (g >= (long long)n * 16) return;
    int i  = (int)(g >> 4);
    int f4 = (int)(g & 15) * 4;
    int gr = gid[i];
    float4 v = *(const float4*)(h2 + (size_t)i * D + f4);
    float* o = pooled + (size_t)gr * D + f4;
    atomicAdd(o + 0, v.x);
    atomicAdd(o + 1, v.y);
    atomicAdd(o + 2, v.z);
    atomicAdd(o + 3, v.w);
    if (f4 == 0) atomicAdd(&counts[gr], 1.0f);
}

// out[g,c] = (pooled[g,:]/count[g]) @ Wc[:,c] + bc[c]   ([128,10], tiny)
__global__ void gcn_head(const float* __restrict__ pooled, const float* __restrict__ counts,
                         const float* __restrict__ Wc, const float* __restrict__ bc,
                         float* __restrict__ out, int G, int C) {
    int idx = blockIdx.x * blockDim.x + threadIdx.x;
    if (idx >= G * C) return;
    int g = idx / C;
    int c = idx % C;
    float inv = 1.0f / fmaxf(counts[g], 1.0f);
    float acc = 0.0f;
    for (int k = 0; k < D; ++k) acc += pooled[(size_t)g * D + k] * Wc[k * C + c];
    out[idx] = acc * inv + bc[c];
}

// ---------------------------------------------------------------------------
extern "C" void kernel_launch(void* const* d_in, const int* in_sizes, int n_in,
                              void* d_out, int out_size, void* d_ws, size_t ws_size,
                              hipStream_t stream) {
    const float* h   = (const float*)d_in[0];
    const float* W1  = (const float*)d_in[1];
    const float* b1  = (const float*)d_in[2];
    const float* W2  = (const float*)d_in[3];
    const float* b2  = (const float*)d_in[4];
    const float* Wc  = (const float*)d_in[5];
    const float* bc  = (const float*)d_in[6];
    const int*   src = (const int*)d_in[7];
    const int*   dst = (const int*)d_in[8];
    const int*   gid = (const int*)d_in[9];
    float*       out = (float*)d_out;

    const int N = in_sizes[0] / D;   // 100000
    const int E = in_sizes[7];       // 1600000
    const int C = 10;
    const int G = out_size / C;      // 128

    // workspace carve-out (256B aligned)
    char* ws = (char*)d_ws;
    size_t off = 0;
    auto carve = [&](size_t bytes) {
        void* p = ws + off;
        off = (off + bytes + 255) & ~(size_t)255;
        return p;
    };
    float* nsrc   = (float*)carve((size_t)N * sizeof(float));       // deg_out -> norm_src
    float* ndst   = (float*)carve((size_t)N * sizeof(float));       // deg_in  -> norm_dst
    float* bufA   = (float*)carve((size_t)N * D * sizeof(float));   // agg accumulator
    float* bufB   = (float*)carve((size_t)N * D * sizeof(float));   // scaled feats / layer out
    float* pooled = (float*)carve((size_t)G * D * sizeof(float));
    float* counts = (float*)carve((size_t)G * sizeof(float));
    (void)ws_size; (void)n_in;

    const int TB = 256;
    const int edgeBlocks  = (E + TB - 1) / TB;
    const int nodeBlocks  = (N + TB - 1) / TB;
    long long nt16n = (long long)N * 16;
    long long nt16e = (long long)E * 16;
    const int node16Blocks = (int)((nt16n + TB - 1) / TB);
    const int edge16Blocks = (int)((nt16e + TB - 1) / TB);
    const int gemmBlocks   = (N + 63) / 64;

    // --- degrees & norms ---
    hipMemsetAsync(nsrc, 0, (size_t)N * sizeof(float), stream);
    hipMemsetAsync(ndst, 0, (size_t)N * sizeof(float), stream);
    gcn_degrees<<<edgeBlocks, TB, 0, stream>>>(src, dst, nsrc, ndst, E);
    gcn_norms<<<nodeBlocks, TB, 0, stream>>>(nsrc, ndst, N);

    // --- layer 1 ---
    gcn_rowscale<<<node16Blocks, TB, 0, stream>>>(h, nsrc, bufB, N);       // xs1 = h * norm_src
    hipMemsetAsync(bufA, 0, (size_t)N * D * sizeof(float), stream);
    gcn_scatter<<<edge16Blocks, TB, 0, stream>>>(bufB, src, dst, bufA, E); // agg1
    // bufB = relu(agg1*norm_dst @ W1 + b1) * norm_src   (prescaled input to layer 2)
    gcn_gemm64_wmma<<<gemmBlocks, 128, 0, stream>>>(bufA, W1, b1, ndst, nsrc, bufB, N);

    // --- layer 2 ---
    hipMemsetAsync(bufA, 0, (size_t)N * D * sizeof(float), stream);
    gcn_scatter<<<edge16Blocks, TB, 0, stream>>>(bufB, src, dst, bufA, E); // agg2
    // bufB = h2 = relu(agg2*norm_dst @ W2 + b2)
    gcn_gemm64_wmma<<<gemmBlocks, 128, 0, stream>>>(bufA, W2, b2, ndst, nullptr, bufB, N);

    // --- mean pool + classifier head ---
    hipMemsetAsync(pooled, 0, (size_t)G * D * sizeof(float), stream);
    hipMemsetAsync(counts, 0, (size_t)G * sizeof(float), stream);
    gcn_pool<<<node16Blocks, TB, 0, stream>>>(bufB, gid, pooled, counts, N);
    gcn_head<<<(G * C + TB - 1) / TB, TB, 0, stream>>>(pooled, counts, Wc, bc, out, G, C);
}